// DifferentiableECE_40905268527129
// MI455X (gfx1250) — compile-verified
//
#include <hip/hip_runtime.h>

// ---------------------------------------------------------------------------
// Differentiable ECE on gfx1250 (MI455X), wave32.
//   Pass 1: grid-stride over 16.8M elements, factored Gaussian:
//             w_b = K_b * exp2(c1*p - c2*p^2) * exp2(c3*p)^b   (K_b deferred)
//           Element PAIRS processed with explicit VOP3P packed-f32 inline asm
//           (v_pk_fma_f32 / v_pk_add_f32 / v_pk_mul_f32) so the 30-accumulator
//           chain runs at 2 lanes/instruction. shfl_xor wave reduce; 8x32 LDS
//           tile collapsed with V_WMMA_F32_16X16X4_F32 (ones x B); per-block
//           partials -> d_ws.
//   Pass 2: single block reduces 2048x30 partials, applies K_b, eps,
//           |conf-acc|, writes the scalar.
// ---------------------------------------------------------------------------

typedef __attribute__((ext_vector_type(2))) float v2f;
typedef __attribute__((ext_vector_type(8))) float v8f;
typedef __attribute__((ext_vector_type(4))) float f32x4;
typedef __attribute__((ext_vector_type(4))) int   i32x4;

#define NBINS   10
#define NACC    30          // {sum_w, sum_wp, sum_wl} x 10 bins (K_b deferred)
#define PSTRIDE 32          // padded partial row (cols 30,31 zero)
#define THREADS 256
#define BLOCKS  2048

// log2(e)-folded constants:  -50*log2e, 5*log2e, 10*log2e
#define C_M50L2E  (-72.134752044448170f)
#define C_P5L2E     (7.2134752044448170f)
#define C_P10L2E   (14.426950408889634f)

// Raw hardware exp2: lowers to a single v_exp_f32 (transcendental unit).
__device__ __forceinline__ float hw_exp2(float x) {
    return __builtin_amdgcn_exp2f(x);
}

// ---- guaranteed VOP3P packed-f32 ops (2 elements per instruction) ----
__device__ __forceinline__ v2f pk_fma(v2f a, v2f b, v2f c) {
    v2f d;
    asm("v_pk_fma_f32 %0, %1, %2, %3" : "=v"(d) : "v"(a), "v"(b), "v"(c));
    return d;
}
__device__ __forceinline__ v2f pk_add(v2f a, v2f b) {
    v2f d;
    asm("v_pk_add_f32 %0, %1, %2" : "=v"(d) : "v"(a), "v"(b));
    return d;
}
__device__ __forceinline__ v2f pk_mul(v2f a, v2f b) {
    v2f d;
    asm("v_pk_mul_f32 %0, %1, %2" : "=v"(d) : "v"(a), "v"(b));
    return d;
}

__device__ __forceinline__ float wave_reduce_add(float v) {
#pragma unroll
    for (int m = 16; m >= 1; m >>= 1)
        v += __shfl_xor(v, m, 32);
    return v;
}

// Process two elements at once with packed-f32 math.
__device__ __forceinline__ void accum_pair(v2f p, v2f l, v2f* acc) {
    const v2f km50 = {C_M50L2E, C_M50L2E};
    const v2f kp5  = {C_P5L2E,  C_P5L2E};
    const v2f k10  = {C_P10L2E, C_P10L2E};
    // t = p*(-50*log2e*p + 5*log2e)  ->  g = 2^t = exp(-50p^2+5p)
    v2f t  = pk_mul(p, pk_fma(km50, p, kp5));
    v2f rt = pk_mul(k10, p);          // r = 2^(10*log2e*p) = exp(10p)
    v2f g, r;
    g.x = hw_exp2(t.x);
    g.y = hw_exp2(t.y);
    r.x = hw_exp2(rt.x);
    r.y = hw_exp2(rt.y);
    v2f cur = g;
#pragma unroll
    for (int b = 0; b < NBINS; ++b) {
        acc[3 * b + 0] = pk_add(acc[3 * b + 0], cur);
        acc[3 * b + 1] = pk_fma(cur, p, acc[3 * b + 1]);
        acc[3 * b + 2] = pk_fma(cur, l, acc[3 * b + 2]);
        cur = pk_mul(cur, r);
    }
}

__global__ __launch_bounds__(THREADS)
void ece_partial_kernel(const float* __restrict__ probs,
                        const int* __restrict__ labels,
                        float* __restrict__ partial,
                        int n4) {
    v2f acc2[NACC];
#pragma unroll
    for (int i = 0; i < NACC; ++i) { acc2[i].x = 0.f; acc2[i].y = 0.f; }

    const f32x4* __restrict__ p4 = (const f32x4*)probs;
    const i32x4* __restrict__ l4 = (const i32x4*)labels;

    const int stride = gridDim.x * THREADS;
    for (int i = blockIdx.x * THREADS + threadIdx.x; i < n4; i += stride) {
        // Streaming read-once data: non-temporal 128-bit loads, spare the L2.
        f32x4 pv = __builtin_nontemporal_load(&p4[i]);
        i32x4 lv = __builtin_nontemporal_load(&l4[i]);
        v2f pa, pb, la, lb;
        pa.x = pv.x; pa.y = pv.y;
        pb.x = pv.z; pb.y = pv.w;
        la.x = (float)lv.x; la.y = (float)lv.y;
        lb.x = (float)lv.z; lb.y = (float)lv.w;
        accum_pair(pa, la, acc2);
        accum_pair(pb, lb, acc2);
    }

    // ---- collapse pairs, intra-wave reduction, 8x32 tile in LDS ----
    __shared__ float red[THREADS / 32][PSTRIDE];
    const int lane = threadIdx.x & 31;
    const int wave = threadIdx.x >> 5;
#pragma unroll
    for (int j = 0; j < NACC; ++j) {
        float v = wave_reduce_add(acc2[j].x + acc2[j].y);
        if (lane == 0) red[wave][j] = v;
    }
    if (lane == 0) { red[wave][30] = 0.f; red[wave][31] = 0.f; }
    __syncthreads();

    // ---- wave 0: collapse 8 rows x 32 cols with V_WMMA_F32_16X16X4_F32 ----
    // A = ones(16x4)  =>  D[m][n] = sum_k B[k][n] for every m.
    // B (4x16 f32, 2 VGPRs): VGPR0 = K0 (lanes 0-15) / K2 (lanes 16-31),
    //                        VGPR1 = K1 (lanes 0-15) / K3 (lanes 16-31).
    if (wave == 0) {
        const int half = lane >> 4;
        const int col  = lane & 15;
        v2f a; a.x = 1.f; a.y = 1.f;
        v8f c0 = {}; v8f c1 = {};
#pragma unroll
        for (int r0 = 0; r0 < THREADS / 32; r0 += 4) {
            v2f b0, b1;
            b0.x = red[r0 + 2 * half + 0][col];
            b0.y = red[r0 + 2 * half + 1][col];
            b1.x = red[r0 + 2 * half + 0][16 + col];
            b1.y = red[r0 + 2 * half + 1][16 + col];
            c0 = __builtin_amdgcn_wmma_f32_16x16x4_f32(false, a, false, b0,
                                                       (short)0, c0, false, false);
            c1 = __builtin_amdgcn_wmma_f32_16x16x4_f32(false, a, false, b1,
                                                       (short)0, c1, false, false);
        }
        // D row M=0 lives in VGPR0 of lanes 0..15 (lane index == column).
        if (lane < 16) {
            float* out = partial + blockIdx.x * PSTRIDE;
            out[col]      = c0[0];
            out[16 + col] = c1[0];
        }
    }
}

__global__ __launch_bounds__(THREADS)
void ece_final_kernel(const float* __restrict__ partial,
                      float* __restrict__ out,
                      int nrows) {
    __shared__ float red[8][PSTRIDE];
    __shared__ float colsum[PSTRIDE];
    const int col = threadIdx.x & 31;
    const int rg  = threadIdx.x >> 5;

    float s = 0.f;
    for (int r = rg; r < nrows; r += 8)
        s += partial[r * PSTRIDE + col];
    red[rg][col] = s;
    __syncthreads();

    if (threadIdx.x < 32) {
        float t = 0.f;
#pragma unroll
        for (int g = 0; g < 8; ++g) t += red[g][threadIdx.x];
        colsum[threadIdx.x] = t;
    }
    __syncthreads();

    if (threadIdx.x == 0) {
        float total = 0.f;
#pragma unroll
        for (int b = 0; b < NBINS; ++b) {
            const float cb = 0.05f + 0.1f * (float)b;
            const float Kb = __expf(-50.f * cb * cb);   // deferred Gaussian constant
            const float ws = Kb * colsum[3 * b + 0];
            const float wp = Kb * colsum[3 * b + 1];
            const float wl = Kb * colsum[3 * b + 2];
            const float inv = 1.f / (ws + 1e-8f);
            // mass * |conf - acc| = (ws*inv) * |wp - wl| * inv
            total += (ws * inv) * fabsf((wp - wl) * inv);
        }
        out[0] = total;
    }
}

extern "C" void kernel_launch(void* const* d_in, const int* in_sizes, int n_in,
                              void* d_out, int out_size, void* d_ws, size_t ws_size,
                              hipStream_t stream) {
    (void)n_in; (void)out_size; (void)ws_size;
    const float* probs  = (const float*)d_in[0];
    const int*   labels = (const int*)d_in[1];
    const int n  = in_sizes[0];      // 2048*8192 = 16,777,216 (divisible by 4)
    const int n4 = n >> 2;

    float* partial = (float*)d_ws;   // BLOCKS * PSTRIDE floats = 256 KB

    ece_partial_kernel<<<BLOCKS, THREADS, 0, stream>>>(probs, labels, partial, n4);
    ece_final_kernel<<<1, THREADS, 0, stream>>>(partial, (float*)d_out, BLOCKS);
}